// MaxMin_54683523612736
// MI455X (gfx1250) — compile-verified
//
#include <hip/hip_runtime.h>

// Native vector types so clang's nontemporal builtins lower to
// global_load_b128 / global_store_b128 with TH=NT on gfx1250.
typedef float v4f __attribute__((ext_vector_type(4)));

// Two adjacent pooling windows (8 floats) per thread:
//   input  floats [8t, 8t+8)  -> 2x 16B NT b128 loads
//   output floats [4t, 4t+4)  -> {idx0, min0, idx1, min1}, 1x 16B NT b128 store
__global__ __launch_bounds__(256) void MaxMin_pool4_kernel(
    const float* __restrict__ in, float* __restrict__ out, int npair) {
  int t = blockIdx.x * blockDim.x + threadIdx.x;
  if (t >= npair) return;

  // gfx1250 prefetch path: one global_prefetch_b8 per wave, 32 KiB (1024
  // window-pairs) ahead of the wave's stream, bounds-guarded so it never
  // points past the allocation.
  if (((threadIdx.x & 31u) == 0u) && (t + 1024 < npair)) {
    __builtin_prefetch((const char*)in + (size_t)t * 32u + 32768u, 0, 0);
  }

  const v4f* src = (const v4f*)in + (size_t)t * 2u;
  v4f w0 = __builtin_nontemporal_load(src);
  v4f w1 = __builtin_nontemporal_load(src + 1);

  // First-max-wins argmax (strict '>' matches jnp.argmax tie-breaking).
  float b0 = w0.x, i0 = 0.0f;
  if (w0.y > b0) { b0 = w0.y; i0 = 1.0f; }
  if (w0.z > b0) { b0 = w0.z; i0 = 2.0f; }
  if (w0.w > b0) { b0 = w0.w; i0 = 3.0f; }
  float m0 = fminf(fminf(w0.x, w0.y), fminf(w0.z, w0.w));

  float b1 = w1.x, i1 = 0.0f;
  if (w1.y > b1) { b1 = w1.y; i1 = 1.0f; }
  if (w1.z > b1) { b1 = w1.z; i1 = 2.0f; }
  if (w1.w > b1) { b1 = w1.w; i1 = 3.0f; }
  float m1 = fminf(fminf(w1.x, w1.y), fminf(w1.z, w1.w));

  v4f r;
  r.x = i0;  // out[4t+0] = argmax(window 2t)
  r.y = m0;  // out[4t+1] = min   (window 2t)
  r.z = i1;  // out[4t+2] = argmax(window 2t+1)
  r.w = m1;  // out[4t+3] = min   (window 2t+1)
  __builtin_nontemporal_store(r, (v4f*)out + t);
}

extern "C" void kernel_launch(void* const* d_in, const int* in_sizes, int n_in,
                              void* d_out, int out_size, void* d_ws, size_t ws_size,
                              hipStream_t stream) {
  const float* in = (const float*)d_in[0];
  float* out = (float*)d_out;

  // Window pairs: (C*H)/POOLSIZE/2. Reference shape: 8,388,608 pairs.
  int npair = in_sizes[0] / 8;

  const int threads = 256;  // 8 wave32s per block
  int blocks = (npair + threads - 1) / threads;

  MaxMin_pool4_kernel<<<blocks, threads, 0, stream>>>(in, out, npair);
}